// GroupAttention_43009802502483
// MI455X (gfx1250) — compile-verified
//
#include <hip/hip_runtime.h>
#include <hip/hip_bf16.h>

#define B_ 4
#define S_ 1024
#define D_ 512
#define BS_ (B_ * S_)

typedef __attribute__((ext_vector_type(2))) float v2f;
typedef __attribute__((ext_vector_type(8))) float v8f;

// ---------------- 1. LayerNorm (unbiased var, eps added to std) ----------------
// one block per row, 128 threads x float4
__global__ void ga_ln_kernel(const float* __restrict__ ctx,
                             const float* __restrict__ gamma,
                             const float* __restrict__ beta,
                             float* __restrict__ xn) {
  int row = blockIdx.x;
  int t = threadIdx.x;
  float4 v = ((const float4*)(ctx + (size_t)row * D_))[t];
  __shared__ float red[128];
  red[t] = v.x + v.y + v.z + v.w;
  __syncthreads();
  for (int o = 64; o > 0; o >>= 1) { if (t < o) red[t] += red[t + o]; __syncthreads(); }
  float mu = red[0] * (1.0f / D_);
  __syncthreads();
  float dx = v.x - mu, dy = v.y - mu, dz = v.z - mu, dw = v.w - mu;
  red[t] = dx * dx + dy * dy + dz * dz + dw * dw;
  __syncthreads();
  for (int o = 64; o > 0; o >>= 1) { if (t < o) red[t] += red[t + o]; __syncthreads(); }
  float var = red[0] * (1.0f / (float)(D_ - 1));
  float inv = 1.0f / (sqrtf(var) + 1e-6f);
  float4 g = ((const float4*)gamma)[t];
  float4 bb = ((const float4*)beta)[t];
  float4 o4;
  o4.x = g.x * dx * inv + bb.x;
  o4.y = g.y * dy * inv + bb.y;
  o4.z = g.z * dz * inv + bb.z;
  o4.w = g.w * dw * inv + bb.w;
  ((float4*)(xn + (size_t)row * D_))[t] = o4;
}

// ---------------- 2. WMMA fp32 GEMM: q = xn@Wq + bq ; k = xn@Wk + bk ----------
// One 16x32 output slab per wave (two 16x16 WMMA tiles -> two independent
// accumulator chains; one A fragment feeds both). M = 4096, N = 512+512, K = 512.
// 256 threads = 8 waves/block; slabs = 256 M-tiles * 32 N-slabs = 8192 waves.
__global__ void ga_qk_gemm_kernel(const float* __restrict__ xn,
                                  const float* __restrict__ Wq,
                                  const float* __restrict__ bq,
                                  const float* __restrict__ Wk,
                                  const float* __restrict__ bk,
                                  float* __restrict__ q,
                                  float* __restrict__ k) {
  int wave = threadIdx.x >> 5;
  int lane = threadIdx.x & 31;
  int slab = blockIdx.x * 8 + wave;
  int tm = slab >> 5;   // 0..255  (M tile)
  int tn = slab & 31;   // 0..31   (32-wide N slab across [q|k])
  const float* W; const float* bias; float* out; int colBase;
  if (tn < 16) { W = Wq; bias = bq; out = q; colBase = tn * 32; }
  else         { W = Wk; bias = bk; out = k; colBase = (tn - 16) * 32; }

  int m15   = lane & 15;
  int mrow  = tm * 16 + m15;
  int ncol0 = colBase + m15;        // first 16-column tile
  int ncol1 = ncol0 + 16;           // second 16-column tile
  int klo   = (lane >> 4) * 2;      // lanes 0-15 -> K{0,1}; lanes 16-31 -> K{2,3}
  const float* arow = xn + (size_t)mrow * D_;

  v8f acc0 = {};
  v8f acc1 = {};
  for (int k0 = 0; k0 < D_; k0 += 4) {
    int kk = k0 + klo;
    v2f a = *(const v2f*)(arow + kk);               // A 16x4 fp32 fragment
    const float* w0 = W + (size_t)kk * D_;
    const float* w1 = W + (size_t)(kk + 1) * D_;
    v2f b0, b1;
    b0.x = w0[ncol0]; b0.y = w1[ncol0];             // B 4x16 fp32 fragment (tile 0)
    b1.x = w0[ncol1]; b1.y = w1[ncol1];             // B 4x16 fp32 fragment (tile 1)
    acc0 = __builtin_amdgcn_wmma_f32_16x16x4_f32(
        false, a, false, b0, (short)0, acc0, false, false);
    acc1 = __builtin_amdgcn_wmma_f32_16x16x4_f32(
        false, a, false, b1, (short)0, acc1, false, false);
  }
  float bv0 = bias[ncol0];
  float bv1 = bias[ncol1];
  int rbase = tm * 16 + ((lane >> 4) * 8);          // C/D: VGPR r -> M=r (+8 hi lanes)
#pragma unroll
  for (int r = 0; r < 8; ++r) {
    out[(size_t)(rbase + r) * D_ + ncol0] = acc0[r] + bv0;
    out[(size_t)(rbase + r) * D_ + ncol1] = acc1[r] + bv1;
  }
}

// ---------------- 3. Band scores + 2-way softmax -------------------------------
// one block per (b,s) row; scores[s,s+1]=q[s].k[s+1], scores[s,s-1]=q[s].k[s-1]
__global__ void ga_band_scores_kernel(const float* __restrict__ q,
                                      const float* __restrict__ k,
                                      const int* __restrict__ eos,
                                      float* __restrict__ attn_next,
                                      float* __restrict__ attn_prev) {
  int row = blockIdx.x;            // b*S + s
  int b = row >> 10;
  int s = row & (S_ - 1);
  int t = threadIdx.x;             // 128 threads x float4
  float4 qa = ((const float4*)(q + (size_t)row * D_))[t];
  float sn = 0.f, sp = 0.f;
  if (s < S_ - 1) {
    float4 kn = ((const float4*)(k + (size_t)(row + 1) * D_))[t];
    sn = qa.x * kn.x + qa.y * kn.y + qa.z * kn.z + qa.w * kn.w;
  }
  if (s > 0) {
    float4 kp = ((const float4*)(k + (size_t)(row - 1) * D_))[t];
    sp = qa.x * kp.x + qa.y * kp.y + qa.z * kp.z + qa.w * kp.w;
  }
  __shared__ float rn[128], rp[128];
  rn[t] = sn; rp[t] = sp;
  __syncthreads();
  for (int o = 64; o > 0; o >>= 1) {
    if (t < o) { rn[t] += rn[t + o]; rp[t] += rp[t + o]; }
    __syncthreads();
  }
  if (t == 0) {
    // NOTE: reference hardcodes scale 256.0
    float en = (s < S_ - 1 && eos[b * S_ + s + 1] != 0) ? rn[0] * (1.0f / 256.0f) : -1e9f;
    float ep = (s > 0      && eos[b * S_ + s - 1] != 0) ? rp[0] * (1.0f / 256.0f) : -1e9f;
    float m = fmaxf(en, ep);
    float wn = expf(en - m), wp = expf(ep - m);
    float z = wn + wp;
    attn_next[row] = wn / z;
    attn_prev[row] = wp / z;
  }
}

// ---------------- 4. Band blend with prior + log -------------------------------
__global__ void ga_band_kernel(const float* __restrict__ prior,
                               const float* __restrict__ attn_next,
                               const float* __restrict__ attn_prev,
                               float* __restrict__ band_up,
                               float* __restrict__ band_dn,
                               float* __restrict__ Lg) {
  int idx = blockIdx.x * blockDim.x + threadIdx.x;
  if (idx >= B_ * (S_ - 1)) return;
  int b = idx / (S_ - 1);
  int s = idx % (S_ - 1);
  int row = b * S_ + s;
  float v = sqrtf(attn_next[row] * attn_prev[row + 1] + 1e-9f);
  size_t pbase = (size_t)b * S_ * S_;
  float pu = prior[pbase + (size_t)s * S_ + (s + 1)];
  float pd = prior[pbase + (size_t)(s + 1) * S_ + s];
  float nu = pu + (1.0f - pu) * v;     // neibor[s, s+1]
  float nd = pd + (1.0f - pd) * v;     // neibor[s+1, s]
  band_up[row] = nu;
  band_dn[row] = nd;
  Lg[row] = logf(nu + 1e-9f);
}

// ---------------- 5. Per-batch exclusive prefix scan of L ----------------------
__global__ void ga_scan_kernel(const float* __restrict__ Lg, float* __restrict__ PS) {
  int b = blockIdx.x;
  int t = threadIdx.x;   // 1024
  __shared__ float sc[1024];
  sc[t] = (t < S_ - 1) ? Lg[b * S_ + t] : 0.0f;
  __syncthreads();
  for (int o = 1; o < 1024; o <<= 1) {
    float add = (t >= o) ? sc[t - o] : 0.0f;
    __syncthreads();
    sc[t] += add;
    __syncthreads();
  }
  PS[b * S_ + t] = (t == 0) ? 0.0f : sc[t - 1];
}

// ---------------- 6. Finalize (bandwidth-bound, float4) ------------------------
// g[i,j] = exp(PS[max]-PS[min]) + 1e-9 (off-diag), neibor diag (on diag)
// neibor = band on |i-j|==1, else prior + (1-prior)*sqrt(1e-9)
__global__ void ga_finalize_kernel(const float* __restrict__ prior,
                                   const float* __restrict__ PS,
                                   const float* __restrict__ band_up,
                                   const float* __restrict__ band_dn,
                                   float* __restrict__ g_out,
                                   float* __restrict__ na_out) {
  const float SQ9 = 3.1622776601683795e-5f;   // sqrt(1e-9)
  int tid = blockIdx.x * blockDim.x + threadIdx.x;   // B*S*S/4 threads
  int jq = tid & (S_ / 4 - 1);
  int i  = (tid >> 8) & (S_ - 1);
  int b  = tid >> 18;
  int j0 = jq * 4;
  size_t base = ((size_t)b * S_ + i) * S_ + j0;
  float4 p4 = *(const float4*)(prior + base);
  const float* PSb = PS + b * S_;
  float psi = PSb[i];
  float4 ps4 = *(const float4*)(PSb + j0);
  float pr[4]  = {p4.x, p4.y, p4.z, p4.w};
  float psj[4] = {ps4.x, ps4.y, ps4.z, ps4.w};
  float na[4], gg[4];
#pragma unroll
  for (int u = 0; u < 4; ++u) {
    int j = j0 + u;
    float p = pr[u];
    float n;
    if (j == i + 1)      n = band_up[b * S_ + i];
    else if (j + 1 == i) n = band_dn[b * S_ + j];
    else                 n = p + (1.0f - p) * SQ9;
    na[u] = n;
    if (i == j)      gg[u] = n;                          // g diag = neibor diag
    else if (i < j)  gg[u] = expf(psj[u] - psi) + 1e-9f;
    else             gg[u] = expf(psi - psj[u]) + 1e-9f;
  }
  *(float4*)(g_out + base)  = make_float4(gg[0], gg[1], gg[2], gg[3]);
  *(float4*)(na_out + base) = make_float4(na[0], na[1], na[2], na[3]);
}

extern "C" void kernel_launch(void* const* d_in, const int* in_sizes, int n_in,
                              void* d_out, int out_size, void* d_ws, size_t ws_size,
                              hipStream_t stream) {
  (void)in_sizes; (void)n_in; (void)out_size; (void)ws_size;
  const float* ctx   = (const float*)d_in[0];
  const float* prior = (const float*)d_in[1];
  const float* Wq    = (const float*)d_in[2];
  const float* bq    = (const float*)d_in[3];
  const float* Wk    = (const float*)d_in[4];
  const float* bk    = (const float*)d_in[5];
  const float* gamma = (const float*)d_in[6];
  const float* beta  = (const float*)d_in[7];
  const int*   eos   = (const int*)d_in[8];

  float* ws = (float*)d_ws;
  float* xn        = ws;                       // 4096*512
  float* q         = xn + (size_t)BS_ * D_;    // 4096*512
  float* k         = q  + (size_t)BS_ * D_;    // 4096*512
  float* attn_next = k  + (size_t)BS_ * D_;    // 4096
  float* attn_prev = attn_next + BS_;          // 4096
  float* band_up   = attn_prev + BS_;          // 4096
  float* band_dn   = band_up + BS_;            // 4096
  float* Lg        = band_dn + BS_;            // 4096
  float* PS        = Lg + BS_;                 // 4096

  float* g_out  = (float*)d_out;                       // B*S*S
  float* na_out = g_out + (size_t)B_ * S_ * S_;        // B*S*S

  ga_ln_kernel<<<BS_, 128, 0, stream>>>(ctx, gamma, beta, xn);
  ga_qk_gemm_kernel<<<(256 * 32) / 8, 256, 0, stream>>>(xn, Wq, bq, Wk, bk, q, k);
  ga_band_scores_kernel<<<BS_, 128, 0, stream>>>(q, k, eos, attn_next, attn_prev);
  ga_band_kernel<<<(B_ * (S_ - 1) + 255) / 256, 256, 0, stream>>>(
      prior, attn_next, attn_prev, band_up, band_dn, Lg);
  ga_scan_kernel<<<B_, 1024, 0, stream>>>(Lg, PS);
  ga_finalize_kernel<<<(B_ * S_ * S_ / 4) / 256, 256, 0, stream>>>(
      prior, PS, band_up, band_dn, g_out, na_out);
}